// LTI_27101243638414
// MI455X (gfx1250) — compile-verified
//
#include <hip/hip_runtime.h>

// ---------------------------------------------------------------------------
// LTI rational filter layer, MI455X (gfx1250, wave32).
//   y = irfft_{2L}( rfft_{2L}(x) * [h0 + B(z)/A(z)] )[:, :L, :]
// Stage 1: 2048-pt radix-2 LDS FFT of x per (b,i), async-scattered load
// Stage 2: per-frequency complex GEMM (v_wmma_f32_16x16x4_f32) with fused
//          transfer-function evaluation; X_k tile async-staged to LDS and
//          overlapped with the VALU-heavy H-tile computation
// Stage 3: 2048-pt inverse FFT per (b,j), async Hermitian-extended load
// Workspace: 4 * 1025*32*256 floats = 134.3 MB (Xr,Xi,Yr,Yi).
// ---------------------------------------------------------------------------

typedef __attribute__((ext_vector_type(2))) float v2f;
typedef __attribute__((ext_vector_type(8))) float v8f;

#define DCH   256        // D_IN == D_OUT
#define BB    32         // batch
#define LL    1024       // sequence length
#define NF2   2048       // padded FFT length
#define NFREQ 1025       // rfft bins of 2048
#define PI_F  3.14159265358979323846f

__device__ __forceinline__ unsigned brev11(unsigned v) { return __brev(v) >> 21; }

// generic (flat) shared pointer -> LDS byte offset (addr[31:0] is LDS offset)
__device__ __forceinline__ unsigned lds_off(const void* p) {
  return (unsigned)(unsigned long long)p;
}

__device__ __forceinline__ void async_b32(unsigned dst_lds, const void* src) {
  asm volatile("global_load_async_to_lds_b32 %0, %1, off"
               :: "v"(dst_lds), "v"(src) : "memory");
}
__device__ __forceinline__ void async_b128(unsigned dst_lds, const void* src) {
  asm volatile("global_load_async_to_lds_b128 %0, %1, off"
               :: "v"(dst_lds), "v"(src) : "memory");
}
__device__ __forceinline__ void wait_async0() {
  asm volatile("s_wait_asynccnt 0x0" ::: "memory");
}

// ---------------------------------------------------------------------------
// Stage 1: forward FFT of x (zero-padded to 2048). One block per (b, i).
// Async loads scatter x directly into bit-reversed LDS slots (even indices);
// zero-fill (odd indices + imag plane) proceeds concurrently on DS path.
// ---------------------------------------------------------------------------
__global__ void __launch_bounds__(256)
fft_fwd_x(const float* __restrict__ x, float* __restrict__ Xr, float* __restrict__ Xi) {
  __shared__ float re[NF2];
  __shared__ float im[NF2];
  const int blk = blockIdx.x;
  const int b   = blk >> 8;
  const int ci  = blk & 255;
  const int tid = threadIdx.x;

  for (int t = tid; t < NF2; t += 256) {
    unsigned r = brev11((unsigned)t);
    im[r] = 0.f;
    if (t >= LL) {
      re[r] = 0.f;                               // t>=L -> odd bit-reversed slot
    } else {
      async_b32(lds_off(&re[r]), x + ((size_t)b * LL + t) * DCH + ci);
    }
  }
  wait_async0();
  __syncthreads();

  for (int s = 0; s < 11; ++s) {
    int half = 1 << s;
    for (int bf = tid; bf < NF2 / 2; bf += 256) {
      int pos = bf & (half - 1);
      int i0  = ((bf >> s) << (s + 1)) | pos;
      int i1  = i0 + half;
      float ang = -PI_F * (float)pos / (float)half;   // forward: e^{-i...}
      float sv, cv;
      __sincosf(ang, &sv, &cv);
      float xr = re[i1], xi = im[i1];
      float tr = cv * xr - sv * xi;
      float ti = cv * xi + sv * xr;
      float ur = re[i0], ui = im[i0];
      re[i1] = ur - tr; im[i1] = ui - ti;
      re[i0] = ur + tr; im[i0] = ui + ti;
    }
    __syncthreads();
  }

  // store bins 0..1024, frequency-major layout [k][b][i]
  for (int k = tid; k < NFREQ; k += 256) {
    size_t o = (size_t)k * (BB * DCH) + b * DCH + ci;
    Xr[o] = re[k];
    Xi[o] = im[k];
  }
}

// ---------------------------------------------------------------------------
// Stage 2: per-frequency complex GEMM with fused transfer-function evaluation.
// Grid: (1025 freqs, 8 j-tiles of 32). Block: 128 threads = 4 waves.
// LDS (128 KB): H tile (256x32 complex) + X_k tile (32x256 complex).
// X tile is async-copied while the H tile is evaluated (copy/compute overlap).
// ---------------------------------------------------------------------------
__global__ void __launch_bounds__(128)
freq_gemm(const float* __restrict__ h0,
          const float* __restrict__ num,     // numerators[0], [i][j][8]
          const float* __restrict__ den,     // denumerator,   [i][j][8]
          const float* __restrict__ Xr, const float* __restrict__ Xi,
          float* __restrict__ Yr, float* __restrict__ Yi) {
  __shared__ float smem[32768];              // 128 KB
  float* sHr = smem;                         // [256][32]
  float* sHi = smem + 8192;                  // [256][32]
  float* sXr = smem + 16384;                 // [32][256]
  float* sXi = smem + 24576;                 // [32][256]

  const int k     = blockIdx.x;
  const int jbase = blockIdx.y * 32;
  const int tid   = threadIdx.x;

  // ---- Phase 0: issue async copy of the X_k tile (64 KB) -------------------
  {
    const float* gXr = Xr + (size_t)k * (BB * DCH);
    const float* gXi = Xi + (size_t)k * (BB * DCH);
    for (int c = tid; c < (BB * DCH) / 4; c += 128) {
      async_b128(lds_off(&sXr[c * 4]), gXr + c * 4);
      async_b128(lds_off(&sXi[c * 4]), gXi + c * 4);
    }
  }

  // Shared twiddles for this bin: z^{-m} = cos(m*theta) - i sin(m*theta)
  const float theta = PI_F * (float)k / 1024.f;
  float cm[8], sm[8];
#pragma unroll
  for (int m = 0; m < 8; ++m) __sincosf(theta * (float)(m + 1), &sm[m], &cm[m]);

  // ---- Phase 1: H tile = h0 + B(z)/A(z), 256 x 32 into LDS -----------------
  for (int e = tid; e < DCH * 32; e += 128) {
    int i  = e >> 5;
    int j  = e & 31;
    int jg = jbase + j;
    const float4* dp = reinterpret_cast<const float4*>(den + ((((size_t)i << 8) + jg) << 3));
    const float4* np = reinterpret_cast<const float4*>(num + ((((size_t)i << 8) + jg) << 3));
    float4 d0 = dp[0], d1 = dp[1];
    float4 n0 = np[0], n1 = np[1];
    float dc[8] = {d0.x, d0.y, d0.z, d0.w, d1.x, d1.y, d1.z, d1.w};
    float nc[8] = {n0.x, n0.y, n0.z, n0.w, n1.x, n1.y, n1.z, n1.w};
    float ar = 1.f, ai = 0.f, br = 0.f, bi = 0.f;
#pragma unroll
    for (int m = 0; m < 8; ++m) {
      ar = fmaf(dc[m], cm[m], ar);
      ai = fmaf(-dc[m], sm[m], ai);
      br = fmaf(nc[m], cm[m], br);
      bi = fmaf(-nc[m], sm[m], bi);
    }
    float inv  = __frcp_rn(fmaf(ar, ar, ai * ai));
    float h0v  = h0[((size_t)i << 8) + jg];
    sHr[e] = fmaf(fmaf(br, ar, bi * ai), inv, h0v);
    sHi[e] = (bi * ar - br * ai) * inv;
  }
  wait_async0();
  __syncthreads();

  // ---- Phase 2: complex GEMM, v_wmma_f32_16x16x4_f32 -----------------------
  // A 16x4 layout: lane&15 = M, lane>>4 selects K pair {0,1}/{2,3}
  // B 4x16 layout: lane&15 = N, lane>>4 selects K pair
  const int lane    = tid & 31;
  const int wave    = tid >> 5;
  const int rowtile = wave & 1;
  const int coltile = wave >> 1;
  const int m16     = lane & 15;
  const int kb      = (lane >> 4) << 1;           // 0 or 2
  const int brow    = rowtile * 16 + m16;         // A-matrix row (batch index)
  const int jj      = coltile * 16 + m16;         // B-matrix column within tile

  const float* xrp = sXr + brow * DCH;
  const float* xip = sXi + brow * DCH;

  v8f accR = {};
  v8f accI = {};
  for (int i0 = 0; i0 < DCH; i0 += 4) {
    int ia = i0 + kb;
    float2 t0 = *reinterpret_cast<const float2*>(xrp + ia);
    float2 t1 = *reinterpret_cast<const float2*>(xip + ia);
    v2f Ar;  Ar.x = t0.x;  Ar.y = t0.y;
    v2f Ai;  Ai.x = t1.x;  Ai.y = t1.y;
    v2f AiN = -Ai;                                 // for Yr -= Xi*Hi
    v2f Br;  Br.x = sHr[ia * 32 + jj];  Br.y = sHr[(ia + 1) * 32 + jj];
    v2f Bi;  Bi.x = sHi[ia * 32 + jj];  Bi.y = sHi[(ia + 1) * 32 + jj];

    accR = __builtin_amdgcn_wmma_f32_16x16x4_f32(false, Ar,  false, Br, (short)0, accR, false, false);
    accR = __builtin_amdgcn_wmma_f32_16x16x4_f32(false, AiN, false, Bi, (short)0, accR, false, false);
    accI = __builtin_amdgcn_wmma_f32_16x16x4_f32(false, Ar,  false, Bi, (short)0, accI, false, false);
    accI = __builtin_amdgcn_wmma_f32_16x16x4_f32(false, Ai,  false, Br, (short)0, accI, false, false);
  }

  // ---- Phase 3: store C tiles (row = vgpr + 8*(lane>=16), col = lane&15) ---
  const int halfsel = lane >> 4;
  const int jo      = jbase + coltile * 16 + m16;
#pragma unroll
  for (int r = 0; r < 8; ++r) {
    int bo   = rowtile * 16 + r + 8 * halfsel;
    size_t o = (size_t)k * (BB * DCH) + bo * DCH + jo;
    Yr[o] = accR[r];
    Yi[o] = accI[r];
  }
}

// ---------------------------------------------------------------------------
// Stage 3: inverse FFT per (b, j) with Hermitian extension; keep t < L.
// Real plane async-scattered into bit-reversed LDS; imag plane needs the
// conjugation sign so it goes through VGPRs.
// ---------------------------------------------------------------------------
__global__ void __launch_bounds__(256)
fft_inv_y(const float* __restrict__ Yr, const float* __restrict__ Yi,
          float* __restrict__ y) {
  __shared__ float re[NF2];
  __shared__ float im[NF2];
  const int blk = blockIdx.x;
  const int b   = blk >> 8;
  const int j   = blk & 255;
  const int tid = threadIdx.x;

  for (int t = tid; t < NF2; t += 256) {
    int   kk  = (t <= LL) ? t : (NF2 - t);
    float sgn = (t <= LL) ? 1.f : -1.f;            // conjugate upper half
    size_t o  = (size_t)kk * (BB * DCH) + b * DCH + j;
    unsigned r = brev11((unsigned)t);
    async_b32(lds_off(&re[r]), Yr + o);
    im[r] = sgn * Yi[o];
  }
  wait_async0();
  __syncthreads();

  for (int s = 0; s < 11; ++s) {
    int half = 1 << s;
    for (int bf = tid; bf < NF2 / 2; bf += 256) {
      int pos = bf & (half - 1);
      int i0  = ((bf >> s) << (s + 1)) | pos;
      int i1  = i0 + half;
      float ang = PI_F * (float)pos / (float)half;  // inverse: e^{+i...}
      float sv, cv;
      __sincosf(ang, &sv, &cv);
      float xr = re[i1], xi = im[i1];
      float tr = cv * xr - sv * xi;
      float ti = cv * xi + sv * xr;
      float ur = re[i0], ui = im[i0];
      re[i1] = ur - tr; im[i1] = ui - ti;
      re[i0] = ur + tr; im[i0] = ui + ti;
    }
    __syncthreads();
  }

  const float scale = 1.f / (float)NF2;
  for (int t = tid; t < LL; t += 256) {
    y[((size_t)b * LL + t) * DCH + j] = re[t] * scale;
  }
}

// ---------------------------------------------------------------------------
extern "C" void kernel_launch(void* const* d_in, const int* in_sizes, int n_in,
                              void* d_out, int out_size, void* d_ws, size_t ws_size,
                              hipStream_t stream) {
  const float* x   = (const float*)d_in[0];   // [32,1024,256]
  const float* h0  = (const float*)d_in[1];   // [256,256,1]
  const float* num = (const float*)d_in[2];   // [2,256,256,8]; causal part = [0]
  const float* den = (const float*)d_in[3];   // [256,256,8]

  float* ws = (float*)d_ws;
  const size_t NSPEC = (size_t)NFREQ * BB * DCH;   // 8,396,800 floats per plane
  float* Xr = ws;
  float* Xi = ws + NSPEC;
  float* Yr = ws + 2 * NSPEC;
  float* Yi = ws + 3 * NSPEC;

  fft_fwd_x<<<dim3(BB * DCH), dim3(256), 0, stream>>>(x, Xr, Xi);
  freq_gemm<<<dim3(NFREQ, 8), dim3(128), 0, stream>>>(h0, num, den, Xr, Xi, Yr, Yi);
  fft_inv_y<<<dim3(BB * DCH), dim3(256), 0, stream>>>(Yr, Yi, (float*)d_out);
}